// RasterizePointsXYsBlending_86363202388522
// MI455X (gfx1250) — compile-verified
//
#include <hip/hip_runtime.h>
#include <stdint.h>

typedef __attribute__((ext_vector_type(2))) float v2f;
typedef __attribute__((ext_vector_type(8))) float v8f;

#define TILE    16
#define KSEL    8
#define MAXC    64      // max candidate points per tile (E[X]~6, Poisson tail ~0)
#define WSTRIDE 66      // even: v2f (b64) A-fragment loads stay 8B aligned
#define FSTRIDE 66      // channel-major feature tile, even for b64 B loads
#define DSTRIDE 260     // D staged [c][pix]; multiple of 4 -> 16B-aligned b128 rows

__global__ __launch_bounds__(256) void rasterize_blend_kernel(
    const float* __restrict__ pts,   // [B, P, 3]
    const float* __restrict__ src,   // [B, 64, P]
    float* __restrict__ out,         // [B, 64, 256, 256]
    int P)
{
    const int tx = blockIdx.x, ty = blockIdx.y, b = blockIdx.z;
    const int t  = threadIdx.x;
    const int H = 256, W = 256, C = 64;

    const float r_ndc  = 1.5f / 256.f * 2.f;       // RADIUS/SIZE*2
    const float r2     = r_ndc * r_ndc;
    const float inv_r2 = 1.f / r2;                 // RAD_POW == 2

    __shared__ __attribute__((aligned(16))) float sW[256 * WSTRIDE]; // W, reused as D[c][pix]
    __shared__ __attribute__((aligned(16))) float sF[C * FSTRIDE];   // F[c][cand]
    __shared__ float sCX[MAXC], sCY[MAXC], sCZ[MAXC];
    __shared__ int   sCI[MAXC];
    __shared__ int   sN;

    const int w0 = tx * TILE, h0 = ty * TILE;
    // NDC extent of this tile's pixel centers (xf = 1-(2w+1)/256, decreasing in w)
    const float xhi = 1.f - (2.f * (float)w0 + 1.f) * (1.f / 256.f) + r_ndc;
    const float xlo = 1.f - (2.f * (float)(w0 + 15) + 1.f) * (1.f / 256.f) - r_ndc;
    const float yhi = 1.f - (2.f * (float)h0 + 1.f) * (1.f / 256.f) + r_ndc;
    const float ylo = 1.f - (2.f * (float)(h0 + 15) + 1.f) * (1.f / 256.f) - r_ndc;

    // ---- 1) deterministic ordered candidate compaction (wave 0) ----
    if (t < 32) {
        const int lane = t;
        int base = 0;
        const float* pb = pts + (size_t)b * P * 3;
        for (int chunk = 0; chunk < P; chunk += 32) {
            const int p = chunk + lane;
            const float x = -pb[p * 3 + 0];        // module negates x,y
            const float y = -pb[p * 3 + 1];
            const float z =  pb[p * 3 + 2];
            const bool ok = (z > 0.f) & (x >= xlo) & (x <= xhi) &
                            (y >= ylo) & (y <= yhi);
            const unsigned msk = (unsigned)__ballot(ok);
            const int pos = base + __popc(msk & ((1u << lane) - 1u));
            if (ok && pos < MAXC) {
                sCX[pos] = x; sCY[pos] = y; sCZ[pos] = z; sCI[pos] = p;
            }
            base += __popc(msk);
        }
        if (lane == 0) sN = (base > MAXC) ? MAXC : base;
    }
    __syncthreads();

    const int ncand  = sN;
    const int ncand4 = (ncand + 3) & ~3;           // pad K-dim for WMMA k=4 steps

    // ---- 2) per-pixel top-K selection + composite weights (dense rows) ----
    {
        const int pyL = t >> 4, pxL = t & 15;
        const float xf = 1.f - (2.f * (float)(w0 + pxL) + 1.f) * (1.f / 256.f);
        const float yf = 1.f - (2.f * (float)(h0 + pyL) + 1.f) * (1.f / 256.f);

        for (int k = 0; k < ncand4; ++k) sW[t * WSTRIDE + k] = 0.f;

        float zs[KSEL], dsd[KSEL]; int js[KSEL];
#pragma unroll
        for (int s = 0; s < KSEL; ++s) { zs[s] = 3.0e38f; dsd[s] = 0.f; js[s] = 0; }

        for (int j = 0; j < ncand; ++j) {
            const float dx = xf - sCX[j];
            const float dy = yf - sCY[j];
            const float d2 = dx * dx + dy * dy;
            if (d2 <= r2) {
                float cz = sCZ[j], cd = d2; int cj = j;
#pragma unroll
                for (int s = 0; s < KSEL; ++s) {   // stable sorted-insert by z
                    const bool  sw = cz < zs[s];
                    const float tz = zs[s], td = dsd[s];
                    const int   tj = js[s];
                    if (sw) { zs[s] = cz; dsd[s] = cd; js[s] = cj;
                              cz = tz; cd = td; cj = tj; }
                }
            }
        }

        float T = 1.f;                              // transmittance
#pragma unroll
        for (int s = 0; s < KSEL; ++s) {
            const bool valid = zs[s] < 1.0e30f;
            float dn = fminf(fmaxf(dsd[s] * inv_r2, 0.001f), 1.f);
            float a  = valid ? (1.f - sqrtf(dn)) : 0.f;   // TAU == 1
            const float wgt = a * T;
            T *= (1.f - a);
            if (valid) sW[t * WSTRIDE + js[s]] = wgt;
        }
    }

    // ---- 3) gather feature tile F[c][cand], channel-major (zero K-pad cols) ----
    for (int e = t; e < ncand4 * C; e += 256) {
        const int c = e & 63, j = e >> 6;
        float v = 0.f;
        if (j < ncand) v = src[((size_t)b * C + c) * (size_t)P + sCI[j]];
        sF[c * FSTRIDE + j] = v;
    }
    __syncthreads();

    // ---- 4) blend GEMM: D[16pix x 16ch] tiles via V_WMMA_F32_16X16X4_F32 ----
    const int lane   = t & 31;
    const int waveId = t >> 5;
    const int half   = lane >> 4;                  // 0: K0/K1, 1: K2/K3
    const int m16    = lane & 15;

    v8f zero = {};
    v8f acc[2][4];
#pragma unroll
    for (int pg = 0; pg < 2; ++pg)
#pragma unroll
        for (int cg = 0; cg < 4; ++cg) acc[pg][cg] = zero;

#pragma unroll
    for (int pg = 0; pg < 2; ++pg) {
        const int rowA = (waveId * 2 + pg) * 16 + m16;     // pixel (A-matrix M)
        for (int k = 0; k < ncand4; k += 4) {
            const int kk = k + 2 * half;
            const v2f a = *(const v2f*)&sW[rowA * WSTRIDE + kk];   // A: {K=kk,kk+1}
#pragma unroll
            for (int cg = 0; cg < 4; ++cg) {
                const v2f bm =                                      // B: {K=kk,kk+1}
                    *(const v2f*)&sF[(cg * 16 + m16) * FSTRIDE + kk];
                acc[pg][cg] = __builtin_amdgcn_wmma_f32_16x16x4_f32(
                    false, a, false, bm, (short)0, acc[pg][cg], false, false);
            }
        }
    }
    __syncthreads();   // all waves done reading sW/sF

    // ---- 5) stage D channel-major in LDS (reuse sW as D[c][pix]) ----
#pragma unroll
    for (int pg = 0; pg < 2; ++pg)
#pragma unroll
        for (int cg = 0; cg < 4; ++cg)
#pragma unroll
            for (int r = 0; r < 8; ++r) {
                const int pix = (waveId * 2 + pg) * 16 + r + 8 * half;
                sW[(cg * 16 + m16) * DSTRIDE + pix] = acc[pg][cg][r];
            }
    __syncthreads();

    // ---- 6) async LDS->global write-out: one b128 per (c, py, 4px) ----
    for (int e = t; e < C * 64; e += 256) {        // 4096 b128 stores per tile
        const int c = e >> 6, pr = e & 63;
        const int py = pr >> 2, pxg = pr & 3;
        const unsigned goff =
            (((unsigned)(b * C + c) * 256u + (unsigned)(h0 + py)) * 256u +
             (unsigned)(w0 + pxg * 4)) * 4u;
        const unsigned lofs =
            (unsigned)(uintptr_t)&sW[c * DSTRIDE + py * TILE + pxg * 4];
        asm volatile("global_store_async_from_lds_b128 %0, %1, %2"
                     :: "v"(goff), "v"(lofs), "s"(out)
                     : "memory");
    }
    asm volatile("s_wait_asynccnt 0" ::: "memory");
}

extern "C" void kernel_launch(void* const* d_in, const int* in_sizes, int n_in,
                              void* d_out, int out_size, void* d_ws, size_t ws_size,
                              hipStream_t stream) {
    (void)n_in; (void)out_size; (void)d_ws; (void)ws_size;
    const float* pts = (const float*)d_in[0];   // [2, P, 3]
    const float* src = (const float*)d_in[1];   // [2, 64, P]
    float* out = (float*)d_out;                 // [2, 64, 256, 256]
    const int P = in_sizes[0] / 6;              // B=2, 3 coords per point
    dim3 grid(256 / TILE, 256 / TILE, 2);
    rasterize_blend_kernel<<<grid, 256, 0, stream>>>(pts, src, out, P);
}